// GNNMoleculeClassifier_80539226734865
// MI455X (gfx1250) — compile-verified
//
#include <hip/hip_runtime.h>

typedef __attribute__((ext_vector_type(2))) float v2f;
typedef __attribute__((ext_vector_type(8))) float v8f;

constexpr int NNODES   = 50000;
constexpr int H        = 128;
constexpr int NCLS     = 10;
constexpr int NCLS_PAD = 16;
constexpr int NGRAPHS  = 512;

static __device__ __forceinline__ void atomic_add_f32(float* p, float v) {
    // non-returning fp32 atomic -> global_atomic_add_f32 (STOREcnt, no stall)
    __hip_atomic_fetch_add(p, v, __ATOMIC_RELAXED, __HIP_MEMORY_SCOPE_AGENT);
}

// ---------------------------------------------------------------- degrees ---
__global__ void __launch_bounds__(256) k_init_deg(unsigned* deg, int n) {
    int i = blockIdx.x * blockDim.x + threadIdx.x;
    if (i < n) deg[i] = 1u;                       // self-loop contributes 1
}

__global__ void __launch_bounds__(256) k_deg_edges(const int* __restrict__ dst,
                                                   unsigned* deg, int e) {
    int i = blockIdx.x * blockDim.x + threadIdx.x;
    if (i < e) atomicAdd(&deg[dst[i]], 1u);
}

__global__ void __launch_bounds__(256) k_dinv(const unsigned* __restrict__ deg,
                                              float* __restrict__ dinv, int n) {
    int i = blockIdx.x * blockDim.x + threadIdx.x;
    if (i < n) dinv[i] = rsqrtf((float)deg[i]);   // deg >= 1 always
}

// per-edge symmetric norm, computed once and reused by all 3 layers
__global__ void __launch_bounds__(256) k_norm(const int* __restrict__ src,
                                              const int* __restrict__ dst,
                                              const float* __restrict__ dinv,
                                              float* __restrict__ norm, int e) {
    int i = blockIdx.x * blockDim.x + threadIdx.x;
    if (i < e) norm[i] = dinv[src[i]] * dinv[dst[i]];
}

// ------------------------------------------------------------ FP32 WMMA GEMM
// out[M,N] = (RELU_IN ? relu(A) : A)[M,K] @ W[K,N] (+bias)(+relu)
// N is a compile-time constant (immediate VMEM offsets, no bounds logic).
// One wave computes NT adjacent 16x16 n-tiles for one m-tile: a single A
// fragment (b64/lane) feeds NT WMMAs per K-step.
template <int N, int NT, bool RELU_IN, bool ADD_BIAS, bool RELU_OUT>
__global__ void __launch_bounds__(256) k_gemm_wmma(
        const float* __restrict__ A, const float* __restrict__ W,
        const float* __restrict__ bias, float* __restrict__ out,
        int M, int K) {
    constexpr int NGRP = (N / 16) / NT;           // wave-column groups
    int wid = blockIdx.x * (blockDim.x >> 5) + (threadIdx.x >> 5);
    int mtiles = M >> 4;
    if (wid >= mtiles * NGRP) return;             // wave-uniform exit
    int mt = wid / NGRP;
    int ng = wid - mt * NGRP;
    int lane = threadIdx.x & 31;
    int lo = lane & 15;
    int hi = lane >> 4;
    int row = (mt << 4) + lo;                     // A row for this lane
    int col = ng * (NT * 16) + lo;                // first output column
    int kg  = hi << 1;                            // lane's k-pair {0,2}

    const float* ap = A + (size_t)row * K + kg;
    const float* bp = W + (size_t)kg * N + col;

    v8f acc[NT] = {};

    #pragma unroll 4
    for (int k0 = 0; k0 < K; k0 += 4) {
        v2f a;
        float a0 = ap[0], a1 = ap[1];
        if (RELU_IN) { a0 = fmaxf(a0, 0.0f); a1 = fmaxf(a1, 0.0f); }
        a.x = a0; a.y = a1;
        #pragma unroll
        for (int j = 0; j < NT; ++j) {
            v2f b;
            b.x = bp[j * 16];                     // immediate offsets
            b.y = bp[j * 16 + N];
            acc[j] = __builtin_amdgcn_wmma_f32_16x16x4_f32(
                         false, a, false, b, (short)0, acc[j], false, false);
        }
        ap += 4;
        bp += 4 * N;
    }

    int m0 = (mt << 4) + (hi << 3);               // C/D layout: M = vgpr + 8*hi
    #pragma unroll
    for (int j = 0; j < NT; ++j) {
        int c = col + j * 16;
        float bv = ADD_BIAS ? bias[c] : 0.0f;
        #pragma unroll
        for (int r = 0; r < 8; ++r) {
            float v = acc[j][r] + bv;
            if (RELU_OUT) v = fmaxf(v, 0.0f);
            out[(size_t)(m0 + r) * N + c] = v;
        }
    }
}

// ------------------------------------------- self-loop term + bias (init) ---
// out[n,f] = bias[f] + dinv[n]^2 * t[n,f]  (full write: clears poisoned ws)
__global__ void __launch_bounds__(256) k_selfloop_bias(
        const float* __restrict__ t, const float* __restrict__ dinv,
        const float* __restrict__ bias, float* __restrict__ out, int n) {
    int i = blockIdx.x * blockDim.x + threadIdx.x;   // over n * (H/4) float4s
    if (i >= n * (H / 4)) return;
    int node = i / (H / 4);
    int f4   = i - node * (H / 4);
    float d = dinv[node];
    float w = d * d;
    float4 v = ((const float4*)t)[i];
    float4 b = ((const float4*)bias)[f4];
    float4 o;
    o.x = fmaf(w, v.x, b.x);
    o.y = fmaf(w, v.y, b.y);
    o.z = fmaf(w, v.z, b.z);
    o.w = fmaf(w, v.w, b.w);
    ((float4*)out)[i] = o;
}

// -------------------------------------------------------- edge scatter-add --
// One wave per edge: lane l owns features [4l,4l+4). Feature matrix (25.6MB)
// and edges live in the 192MB L2; fp32 L2 atomics do the reduction.
__global__ void __launch_bounds__(256) k_scatter(
        const int* __restrict__ src, const int* __restrict__ dst,
        const float* __restrict__ norm, const float* __restrict__ t,
        float* __restrict__ out, int E) {
    int e = blockIdx.x * (blockDim.x >> 5) + (threadIdx.x >> 5);
    if (e >= E) return;
    int s = src[e], d = dst[e];
    float w = norm[e];
    int l = threadIdx.x & 31;
    float4 v = ((const float4*)(t + (size_t)s * H))[l];
    float* o = out + (size_t)d * H + (l << 2);
    atomic_add_f32(o + 0, w * v.x);
    atomic_add_f32(o + 1, w * v.y);
    atomic_add_f32(o + 2, w * v.z);
    atomic_add_f32(o + 3, w * v.w);
}

// ------------------------------------------------------------------- pool ---
__global__ void __launch_bounds__(256) k_zero(float* p, int n) {
    int i = blockIdx.x * blockDim.x + threadIdx.x;
    if (i < n) p[i] = 0.0f;
}

// g[batch[n], :] += relu(h[n, :])
__global__ void __launch_bounds__(256) k_pool(
        const float* __restrict__ h, const int* __restrict__ batch,
        float* __restrict__ g, int n) {
    int node = blockIdx.x * (blockDim.x >> 5) + (threadIdx.x >> 5);
    if (node >= n) return;
    int b = batch[node];
    int l = threadIdx.x & 31;
    float4 v = ((const float4*)(h + (size_t)node * H))[l];
    float* o = g + (size_t)b * H + (l << 2);
    atomic_add_f32(o + 0, fmaxf(v.x, 0.0f));
    atomic_add_f32(o + 1, fmaxf(v.y, 0.0f));
    atomic_add_f32(o + 2, fmaxf(v.z, 0.0f));
    atomic_add_f32(o + 3, fmaxf(v.w, 0.0f));
}

// ------------------------------------- head padding / output compaction -----
__global__ void __launch_bounds__(256) k_pad_head(
        const float* __restrict__ W2, const float* __restrict__ b2,
        float* __restrict__ Wp, float* __restrict__ bpad) {
    int i = blockIdx.x * blockDim.x + threadIdx.x;   // over H*16
    if (i < H * NCLS_PAD) {
        int r = i / NCLS_PAD, c = i - r * NCLS_PAD;
        Wp[i] = (c < NCLS) ? W2[r * NCLS + c] : 0.0f;
    }
    if (i < NCLS_PAD) bpad[i] = (i < NCLS) ? b2[i] : 0.0f;
}

__global__ void __launch_bounds__(256) k_compact_out(
        const float* __restrict__ po, float* __restrict__ out) {
    int i = blockIdx.x * blockDim.x + threadIdx.x;   // NGRAPHS*NCLS
    if (i < NGRAPHS * NCLS) {
        int g = i / NCLS, c = i - g * NCLS;
        out[i] = po[g * NCLS_PAD + c];
    }
}

// ------------------------------------------------------------------ driver --
extern "C" void kernel_launch(void* const* d_in, const int* in_sizes, int n_in,
                              void* d_out, int out_size, void* d_ws, size_t ws_size,
                              hipStream_t stream) {
    const float* x    = (const float*)d_in[0];
    const int*   edge = (const int*)d_in[1];
    const int*   batch= (const int*)d_in[2];
    const float* W0 = (const float*)d_in[3];  const float* b0 = (const float*)d_in[4];
    const float* W1 = (const float*)d_in[5];  const float* b1 = (const float*)d_in[6];
    const float* W2 = (const float*)d_in[7];  const float* b2 = (const float*)d_in[8];
    const float* mW0= (const float*)d_in[9];  const float* mb0= (const float*)d_in[10];
    const float* mW1= (const float*)d_in[11]; const float* mb1= (const float*)d_in[12];
    const float* mW2= (const float*)d_in[13]; const float* mb2= (const float*)d_in[14];

    const int N = NNODES;
    const int E = in_sizes[1] / 2;
    const int FIN = in_sizes[0] / N;              // 64
    const int* src = edge;
    const int* dst = edge + E;

    // workspace layout (fp32)
    float* ws = (float*)d_ws;
    size_t NH = (size_t)N * H;
    float*    bufA = ws;                          // GEMM result t
    float*    bufB = bufA + NH;                   // aggregated h (layers 0,2)
    float*    bufC = bufB + NH;                   // aggregated h (layer 1)
    unsigned* deg  = (unsigned*)(bufC + NH);
    float*    dinv = (float*)(deg + N);
    float*    norm = dinv + N;                    // E floats
    float*    g0   = norm + E;                    // pooled [512,128]
    float*    g1   = g0 + (size_t)NGRAPHS * H;
    float*    g2   = g1 + (size_t)NGRAPHS * H;
    float*    Wpad = g2 + (size_t)NGRAPHS * H;    // [128,16]
    float*    bpad = Wpad + (size_t)H * NCLS_PAD; // [16]
    float*    opad = bpad + NCLS_PAD;             // [512,16]

    auto cdiv = [](int a, int b) { return (a + b - 1) / b; };

    // normalization coefficients
    k_init_deg <<<cdiv(N, 256), 256, 0, stream>>>(deg, N);
    k_deg_edges<<<cdiv(E, 256), 256, 0, stream>>>(dst, deg, E);
    k_dinv     <<<cdiv(N, 256), 256, 0, stream>>>(deg, dinv, N);
    k_norm     <<<cdiv(E, 256), 256, 0, stream>>>(src, dst, dinv, norm, E);
    k_pad_head <<<cdiv(H * NCLS_PAD, 256), 256, 0, stream>>>(mW2, mb2, Wpad, bpad);

    // H-wide GEMM launcher: waves = mtiles * (8 n-tiles / 4 per wave)
    auto gemmH = [&](auto kern, const float* Ain, const float* Wm,
                     const float* bias, float* Out, int M, int K) {
        int waves = (M >> 4) * ((H / 16) / 4);
        kern<<<cdiv(waves, 8), 256, 0, stream>>>(Ain, Wm, bias, Out, M, K);
    };
    int initBlocks = cdiv(N * (H / 4), 256);
    int edgeBlocks = cdiv(E, 8);

    // GCN layer 0:  x -> t(A) -> h(B)
    gemmH(k_gemm_wmma<H, 4, false, false, false>, x, W0, nullptr, bufA, N, FIN);
    k_selfloop_bias<<<initBlocks, 256, 0, stream>>>(bufA, dinv, b0, bufB, N);
    k_scatter      <<<edgeBlocks, 256, 0, stream>>>(src, dst, norm, bufA, bufB, E);

    // GCN layer 1:  relu(B) -> t(A) -> h(C)
    gemmH(k_gemm_wmma<H, 4, true, false, false>, bufB, W1, nullptr, bufA, N, H);
    k_selfloop_bias<<<initBlocks, 256, 0, stream>>>(bufA, dinv, b1, bufC, N);
    k_scatter      <<<edgeBlocks, 256, 0, stream>>>(src, dst, norm, bufA, bufC, E);

    // GCN layer 2:  relu(C) -> t(A) -> h(B)
    gemmH(k_gemm_wmma<H, 4, true, false, false>, bufC, W2, nullptr, bufA, N, H);
    k_selfloop_bias<<<initBlocks, 256, 0, stream>>>(bufA, dinv, b2, bufB, N);
    k_scatter      <<<edgeBlocks, 256, 0, stream>>>(src, dst, norm, bufA, bufB, E);

    // global add-pool of relu(h3)
    k_zero<<<cdiv(NGRAPHS * H, 256), 256, 0, stream>>>(g0, NGRAPHS * H);
    k_pool<<<cdiv(N, 8), 256, 0, stream>>>(bufB, batch, g0, N);

    // MLP head (bias + relu fused into GEMM epilogue)
    gemmH(k_gemm_wmma<H, 4, false, true, true>, g0, mW0, mb0, g1, NGRAPHS, H);
    gemmH(k_gemm_wmma<H, 4, false, true, true>, g1, mW1, mb1, g2, NGRAPHS, H);

    // final 128->10 as padded 128->16 (all loads unconditional), then compact
    {
        int waves = (NGRAPHS >> 4) * ((NCLS_PAD / 16) / 1);
        k_gemm_wmma<NCLS_PAD, 1, false, true, false>
            <<<cdiv(waves, 8), 256, 0, stream>>>(g2, Wpad, bpad, opad, NGRAPHS, H);
    }
    k_compact_out<<<cdiv(NGRAPHS * NCLS, 256), 256, 0, stream>>>(opad, (float*)d_out);
}